// Attention_88811333746876
// MI455X (gfx1250) — compile-verified
//
#include <hip/hip_runtime.h>
#include <math.h>

// Problem constants (from reference)
#define B_  32
#define HS_ 512
#define C_  256
#define H_  64
#define W_  64
#define A_  256
#define HW_ (H_ * W_)

typedef __attribute__((ext_vector_type(2))) float v2f;
typedef __attribute__((ext_vector_type(8))) float v8f;

// ---------------------------------------------------------------------------
// Kernel 1: wh = h_dec(32x512) @ W_h(512x256) + b_h   via V_WMMA_F32_16X16X4_F32
// One wave (32 threads) per 16x16 output tile; 2 x 16 tiles.
// ---------------------------------------------------------------------------
__global__ __launch_bounds__(32) void wh_wmma_kernel(
    const float* __restrict__ h_dec, const float* __restrict__ W_h,
    const float* __restrict__ b_h, float* __restrict__ wh)
{
    const int lane   = threadIdx.x;      // 0..31, EXEC all ones
    const int tile_m = blockIdx.x >> 4;  // 0..1  (B rows)
    const int tile_n = blockIdx.x & 15;  // 0..15 (A cols)
    const int half   = lane >> 4;        // selects K pair {0,1} vs {2,3}
    const int l      = lane & 15;
    const int row    = tile_m * 16 + l;  // A-matrix row held by this lane
    const int col    = tile_n * 16 + l;  // B/D column held by this lane

    v8f acc = {};
    for (int k0 = 0; k0 < HS_; k0 += 4) {
        const int ka = k0 + half * 2;
        v2f a, b;
        a.x = h_dec[row * HS_ + ka];
        a.y = h_dec[row * HS_ + ka + 1];
        b.x = W_h[(ka)     * A_ + col];
        b.y = W_h[(ka + 1) * A_ + col];
        acc = __builtin_amdgcn_wmma_f32_16x16x4_f32(
            false, a, false, b, (short)0, acc, false, false);
    }
    const float bias = b_h[col];
#pragma unroll
    for (int r = 0; r < 8; ++r) {
        const int m = tile_m * 16 + r + half * 8;   // VGPR r: rows r / r+8
        wh[m * A_ + col] = acc[r] + bias;
    }
}

// ---------------------------------------------------------------------------
// Kernel 2: v = wh(32x256) @ W_fm^T(256x256)   (v[b,c] = sum_a wh[b,a]*W_fm[c,a])
// Same WMMA tiling; B-operand reads W_fm transposed (row-major (C,A) storage).
// ---------------------------------------------------------------------------
__global__ __launch_bounds__(32) void v_wmma_kernel(
    const float* __restrict__ wh, const float* __restrict__ W_fm,
    float* __restrict__ v)
{
    const int lane   = threadIdx.x;
    const int tile_m = blockIdx.x >> 4;  // 0..1  (B rows)
    const int tile_n = blockIdx.x & 15;  // 0..15 (C cols)
    const int half   = lane >> 4;
    const int l      = lane & 15;
    const int row    = tile_m * 16 + l;
    const int col    = tile_n * 16 + l;

    v8f acc = {};
    for (int k0 = 0; k0 < A_; k0 += 4) {
        const int ka = k0 + half * 2;
        v2f a, b;
        a.x = wh[row * A_ + ka];
        a.y = wh[row * A_ + ka + 1];
        // B[k][n] = W_fm[n*A + k] -> contiguous pair per lane
        b.x = W_fm[col * A_ + ka];
        b.y = W_fm[col * A_ + ka + 1];
        acc = __builtin_amdgcn_wmma_f32_16x16x4_f32(
            false, a, false, b, (short)0, acc, false, false);
    }
#pragma unroll
    for (int r = 0; r < 8; ++r) {
        const int m = tile_m * 16 + r + half * 8;
        v[m * C_ + col] = acc[r];
    }
}

// ---------------------------------------------------------------------------
// Kernel 3: scores[b,p] = sum_c fm[b,c,p] * v[b,c]   (memory bound, float4)
// grid = (4, 32); block = 256 threads; each thread owns 4 consecutive pixels.
// ---------------------------------------------------------------------------
__global__ __launch_bounds__(256) void scores_kernel(
    const float* __restrict__ fm, const float* __restrict__ v,
    float* __restrict__ scores)
{
    const int b = blockIdx.y;
    const int t = threadIdx.x;
    __shared__ float vs[C_];
    vs[t] = v[b * C_ + t];
    __syncthreads();

    const int p0 = blockIdx.x * 1024 + t * 4;
    const float* fb = fm + (size_t)b * C_ * HW_;
    float4 acc = make_float4(0.f, 0.f, 0.f, 0.f);
#pragma unroll 4
    for (int c = 0; c < C_; ++c) {
        const float4 f = *reinterpret_cast<const float4*>(fb + (size_t)c * HW_ + p0);
        const float w = vs[c];
        acc.x += f.x * w; acc.y += f.y * w; acc.z += f.z * w; acc.w += f.w * w;
    }
    *reinterpret_cast<float4*>(scores + b * HW_ + p0) = acc;
}

// ---------------------------------------------------------------------------
// Kernel 4: in-place softmax over the 4096 spatial scores of each batch.
// grid = 32 blocks; block = 256 threads; LDS tree reductions.
// ---------------------------------------------------------------------------
__global__ __launch_bounds__(256) void softmax_kernel(float* __restrict__ scores)
{
    const int b = blockIdx.x;
    const int t = threadIdx.x;
    float* s = scores + b * HW_;
    __shared__ float red[256];

    float m = -INFINITY;
    for (int i = t; i < HW_; i += 256) m = fmaxf(m, s[i]);
    red[t] = m; __syncthreads();
    for (int k = 128; k > 0; k >>= 1) {
        if (t < k) red[t] = fmaxf(red[t], red[t + k]);
        __syncthreads();
    }
    const float mx = red[0];
    __syncthreads();

    float sum = 0.f;
    for (int i = t; i < HW_; i += 256) {
        const float e = expf(s[i] - mx);
        s[i] = e;
        sum += e;
    }
    red[t] = sum; __syncthreads();
    for (int k = 128; k > 0; k >>= 1) {
        if (t < k) red[t] += red[t + k];
        __syncthreads();
    }
    const float inv = 1.0f / red[0];
    __syncthreads();
    for (int i = t; i < HW_; i += 256) s[i] *= inv;
}

// ---------------------------------------------------------------------------
// Kernel 5: out[b,c] = sum_p fm[b,c,p] * normed[b,p]
// grid = (32, 32); block = 256 threads; 8 channels per block; normed map in LDS.
// ---------------------------------------------------------------------------
__global__ __launch_bounds__(256) void context_kernel(
    const float* __restrict__ fm, const float* __restrict__ normed,
    float* __restrict__ out)
{
    const int b  = blockIdx.y;
    const int c0 = blockIdx.x * 8;
    const int t  = threadIdx.x;

    __shared__ float ns[HW_];     // 16 KB (LDS is 320 KB/WGP)
    __shared__ float red[256];
    for (int i = t; i < HW_; i += 256) ns[i] = normed[b * HW_ + i];
    __syncthreads();

    const float* fb = fm + (size_t)b * C_ * HW_;
    for (int cc = 0; cc < 8; ++cc) {
        const int c = c0 + cc;
        const float* fr = fb + (size_t)c * HW_;
        float acc = 0.f;
#pragma unroll
        for (int i = 0; i < 4; ++i) {
            const int p = i * 1024 + t * 4;
            const float4 f = *reinterpret_cast<const float4*>(fr + p);
            acc += f.x * ns[p] + f.y * ns[p + 1] + f.z * ns[p + 2] + f.w * ns[p + 3];
        }
        red[t] = acc; __syncthreads();
        for (int k = 128; k > 0; k >>= 1) {
            if (t < k) red[t] += red[t + k];
            __syncthreads();
        }
        if (t == 0) out[b * C_ + c] = red[0];
        __syncthreads();
    }
}

// ---------------------------------------------------------------------------
extern "C" void kernel_launch(void* const* d_in, const int* in_sizes, int n_in,
                              void* d_out, int out_size, void* d_ws, size_t ws_size,
                              hipStream_t stream)
{
    (void)in_sizes; (void)n_in; (void)out_size; (void)ws_size;
    const float* h_dec = (const float*)d_in[0];   // (B, HS)
    const float* fm    = (const float*)d_in[1];   // (B, C, H, W)
    const float* W_fm  = (const float*)d_in[2];   // (C, A)
    // d_in[3] = b_fm: adds a per-batch constant to scores -> cancels in softmax
    const float* W_h   = (const float*)d_in[4];   // (HS, A)
    const float* b_h   = (const float*)d_in[5];   // (A,)
    float* out = (float*)d_out;                   // (B, C)

    float* ws = (float*)d_ws;
    float* wh = ws;                      // B*A   = 8192 floats
    float* v  = ws + B_ * A_;            // B*C   = 8192 floats
    float* sc = ws + 2 * B_ * A_;        // B*HW  = 131072 floats (scores -> normed)

    wh_wmma_kernel<<<dim3(32), dim3(32), 0, stream>>>(h_dec, W_h, b_h, wh);
    v_wmma_kernel <<<dim3(32), dim3(32), 0, stream>>>(wh, W_fm, v);
    scores_kernel <<<dim3(4, B_), dim3(256), 0, stream>>>(fm, v, sc);
    softmax_kernel<<<dim3(B_), dim3(256), 0, stream>>>(sc);
    context_kernel<<<dim3(C_ / 8, B_), dim3(256), 0, stream>>>(fm, sc, out);
}